// Probabilstic_Surrogate_58454504898834
// MI455X (gfx1250) — compile-verified
//
#include <hip/hip_runtime.h>
#include <hip/hip_bf16.h>

typedef __attribute__((ext_vector_type(2))) float v2f;
typedef __attribute__((ext_vector_type(8))) float v8f;

#define TILE  16
#define KNB   50
#define NMETA (TILE * KNB)   // 800 metadata entries per side per tile
#define D     64

__global__ __launch_bounds__(256) void graphrec_fused_kernel(
    const int* __restrict__ nodes_u, const int* __restrict__ nodes_v,
    const int* __restrict__ u_neibs, const int* __restrict__ v_neibs,
    const unsigned char* __restrict__ u_mask, const unsigned char* __restrict__ v_mask,
    const float* __restrict__ u_sel, const float* __restrict__ v_sel,
    const float* __restrict__ u2e, const float* __restrict__ v2e,
    const float* __restrict__ Wu, const float* __restrict__ bu,
    const float* __restrict__ Wv, const float* __restrict__ bv,
    float* __restrict__ out)
{
    __shared__ float Xu[TILE][2 * D];   // [self | agg]  8 KB
    __shared__ float Xv[TILE][2 * D];   //               8 KB
    __shared__ float Lu[TILE][D];       // GEMM result   4 KB
    __shared__ float Lv[TILE][D];       //               4 KB
    __shared__ int2  MU[NMETA];         // {nid, w}      6.4 KB
    __shared__ int2  MV[NMETA];         //               6.4 KB
    __shared__ float cntU[TILE];
    __shared__ float cntV[TILE];
    __shared__ float red[TILE][16];

    const int tid   = threadIdx.x;
    const int tile0 = blockIdx.x * TILE;

    // ---------------- Phase 0: init degree accumulators --------------------------
    if (tid < TILE) { cntU[tid] = 0.f; cntV[tid] = 0.f; }
    __syncthreads();

    // ---------------- Phase 1: coalesced metadata staging ------------------------
    // Tile's neibs/sel/mask slices are contiguous [tile0*KNB, tile0*KNB+800).
    // Pack {nid, w = sel*mask} once per entry; accumulate degree via ds_add_f32.
    {
        const size_t base = (size_t)tile0 * KNB;
        #pragma unroll 1
        for (int e = tid; e < NMETA; e += 256) {
            const int rr = e / KNB;   // row within tile
            {
                const float m = (float)u_mask[base + e];
                MU[e] = make_int2(u_neibs[base + e],
                                  __float_as_int(u_sel[base + e] * m));
                atomicAdd(&cntU[rr], m);
            }
            {
                const float m = (float)v_mask[base + e];
                MV[e] = make_int2(v_neibs[base + e],
                                  __float_as_int(v_sel[base + e] * m));
                atomicAdd(&cntV[rr], m);
            }
        }
    }
    __syncthreads();

    // ---------------- Phase 2: gather + weighted-mean aggregation ----------------
    // 256 threads = 16 rows x 16 lanes, each lane owns 4 consecutive dims (float4).
    // Hot loop: 1 ds_load_b64 (meta) + 1 global_load_b128 (embedding) per neighbor.
    const int r   = tid >> 4;      // 0..15  batch row within tile
    const int c   = tid & 15;      // 0..15  -> dims 4c..4c+3
    const int row = tile0 + r;

    #pragma unroll 1
    for (int side = 0; side < 2; ++side) {
        const int*   nodes  = side ? nodes_v : nodes_u;
        const float* tab    = side ? v2e     : u2e;
        const int2*  meta   = side ? MV      : MU;
        const float* cnt    = side ? cntV    : cntU;
        float (*X)[2 * D]   = side ? Xv      : Xu;

        // self embedding (coalesced 256B row: 16 lanes x b128)
        const int node  = nodes[row];
        const float4 s  = ((const float4*)(tab + (size_t)node * D))[c];
        X[r][4 * c + 0] = s.x;
        X[r][4 * c + 1] = s.y;
        X[r][4 * c + 2] = s.z;
        X[r][4 * c + 3] = s.w;

        const int2* mrow = meta + r * KNB;
        float4 acc = make_float4(0.f, 0.f, 0.f, 0.f);
        #pragma unroll 5
        for (int k = 0; k < KNB; ++k) {
            const int2  t = mrow[k];                   // ds_load_b64
            const float w = __int_as_float(t.y);
            const float4 e = ((const float4*)(tab + (size_t)t.x * D))[c];
            acc.x = fmaf(w, e.x, acc.x);
            acc.y = fmaf(w, e.y, acc.y);
            acc.z = fmaf(w, e.z, acc.z);
            acc.w = fmaf(w, e.w, acc.w);
        }
        const float inv = 1.0f / fmaxf(cnt[r], 1.0f);
        X[r][D + 4 * c + 0] = acc.x * inv;
        X[r][D + 4 * c + 1] = acc.y * inv;
        X[r][D + 4 * c + 2] = acc.z * inv;
        X[r][D + 4 * c + 3] = acc.w * inv;
    }
    __syncthreads();

    // ---------------- Phase 3: lin = X @ W + b via V_WMMA_F32_16X16X4_F32 --------
    // 8 waves: wave = side*4 + ntile. Each wave: 16x16 output tile, K=128 by 4.
    {
        const int wave  = tid >> 5;
        const int lane  = tid & 31;
        const int side  = wave >> 2;
        const int ntile = wave & 3;

        const float (*X)[2 * D] = side ? Xv : Xu;
        const float* W          = side ? Wv : Wu;
        const float* bias       = side ? bv : bu;
        float (*L)[D]           = side ? Lv : Lu;

        const int m    = lane & 15;            // A row owned by this lane
        const int koff = (lane >> 4) << 1;     // lanes 0-15 -> K={0,1}; 16-31 -> K={2,3}
        const int n    = ntile * 16 + (lane & 15);

        v8f acc = {0.f, 0.f, 0.f, 0.f, 0.f, 0.f, 0.f, 0.f};
        #pragma unroll 4
        for (int k = 0; k < 2 * D; k += 4) {
            v2f a, b;
            a.x = X[m][k + koff];                       // ds_load_2addr_b64
            a.y = X[m][k + koff + 1];
            b.x = W[(size_t)(k + koff)     * D + n];    // L2-resident weights
            b.y = W[(size_t)(k + koff + 1) * D + n];
            acc = __builtin_amdgcn_wmma_f32_16x16x4_f32(
                      /*neg_a=*/false, a, /*neg_b=*/false, b,
                      /*c_mod=*/(short)0, acc, /*reuse_a=*/false, /*reuse_b=*/false);
        }
        // D layout: VGPR i -> (lanes 0-15: M=i, N=lane) (lanes 16-31: M=i+8, N=lane-16)
        const float bn    = bias[n];
        const int   mbase = (lane >> 4) << 3;  // 0 or 8
        #pragma unroll
        for (int i = 0; i < 8; ++i) {
            L[mbase + i][n] = acc[i] + bn;
        }
    }
    __syncthreads();

    // ---------------- Phase 4: deg==0 fallback + dot(U,V) ------------------------
    {
        const float du = cntU[r];
        const float dv = cntV[r];
        float p = 0.f;
        #pragma unroll
        for (int j = 0; j < 4; ++j) {
            const int d    = c + 16 * j;
            const float ue = (du > 0.f) ? Lu[r][d] : Xu[r][d];  // X[:, :64] = self
            const float ve = (dv > 0.f) ? Lv[r][d] : Xv[r][d];
            p = fmaf(ue, ve, p);
        }
        red[r][c] = p;
    }
    __syncthreads();

    if (tid < TILE) {
        float s = 0.f;
        #pragma unroll
        for (int i = 0; i < 16; ++i) s += red[tid][i];
        out[tile0 + tid] = s;
    }
}

extern "C" void kernel_launch(void* const* d_in, const int* in_sizes, int n_in,
                              void* d_out, int out_size, void* d_ws, size_t ws_size,
                              hipStream_t stream) {
    const int*           nodes_u = (const int*)d_in[0];
    const int*           nodes_v = (const int*)d_in[1];
    const int*           u_neibs = (const int*)d_in[2];
    const int*           v_neibs = (const int*)d_in[3];
    const unsigned char* u_mask  = (const unsigned char*)d_in[4];  // jax bool -> 1 byte
    const unsigned char* v_mask  = (const unsigned char*)d_in[5];
    const float*         u_sel   = (const float*)d_in[6];
    const float*         v_sel   = (const float*)d_in[7];
    const float*         u2e     = (const float*)d_in[8];
    const float*         v2e     = (const float*)d_in[9];
    const float*         Wu      = (const float*)d_in[10];
    const float*         bu      = (const float*)d_in[11];
    const float*         Wv      = (const float*)d_in[12];
    const float*         bv      = (const float*)d_in[13];
    float*               out     = (float*)d_out;

    const int batch = in_sizes[0];           // 16384
    const int grid  = batch / TILE;          // 1024 blocks of 256 threads

    graphrec_fused_kernel<<<grid, 256, 0, stream>>>(
        nodes_u, nodes_v, u_neibs, v_neibs, u_mask, v_mask,
        u_sel, v_sel, u2e, v2e, Wu, bu, Wv, bv, out);
}